// TransitionLoss_30666066494151
// MI455X (gfx1250) — compile-verified
//
#include <hip/hip_runtime.h>
#include <hip/hip_bf16.h>
#include <math.h>

// ---------------------------------------------------------------------------
// TransitionLoss for MI455X (gfx1250, wave32).
//  Kernel 1: G = F * F^T. 8-wave workgroup computes a 64x64 block of G.
//            A/B 64x64 f32 tiles double-buffered in LDS via CDNA5 async
//            copies (global_load_async_to_lds_b128 + s_wait_asynccnt),
//            math on the f32 matrix pipe (V_WMMA_F32_16X16X4_F32) to keep
//            the d=2048 dot products in full f32 like the reference.
//  Kernel 2: per-anchor masked softmax + round-trip transition loss staged
//            in CDNA5's 320KB/WGP LDS (two 128x128 f32 tiles).
//  Kernel 3: hard-mining accuracy from G (dist^2 = Gii + Gjj - 2Gij).
//  Kernel 4: final reductions -> d_out[0] = loss, d_out[1] = correct count.
// ---------------------------------------------------------------------------

typedef float v2f __attribute__((ext_vector_type(2)));
typedef float v8f __attribute__((ext_vector_type(8)));

#define MARGIN_MINUS_ALPHA (-0.2f)   // MARGIN_1 - ALPHA_1 = 0.3 - 0.5
#define ALPHA_1F 0.5f
#define EPS_T 1e-12f

#define MAXM 128   // max intra/cross negative list length (actual: 124)
#define MAXP 32    // max positive list length (actual: 4)

#define TILE 64    // G block edge per workgroup
#define KCH  64    // K chunk staged per buffer
#define LDSP 68    // padded LDS row stride in floats (272B = 17 x 16B chunks)

// ---------------------------------------------------------------------------
// Kernel 1: Gram matrix G[NxN] = F[NxD] * F[NxD]^T.
// Grid (N/64, N/64), 256 threads (8 waves). Wave w: tile row (w&3), tile-col
// pair (w>>2) -> two 16x16 WMMA accumulators sharing the A operand.
// f32 WMMA operand layout (16x4): lane m = lane&15 selects the row, VGPR v
// holds K = v + 2*(lane>>4) -> per-lane float2 at column (kk + 2*(lane>>4)).
// ---------------------------------------------------------------------------
__global__ __launch_bounds__(256) void gram_wmma_kernel(const float* __restrict__ F,
                                                        float* __restrict__ G,
                                                        int N, int D) {
  __shared__ __align__(16) float As[2][TILE][LDSP];
  __shared__ __align__(16) float Bs[2][TILE][LDSP];

  const int tid  = threadIdx.x;
  const int lane = tid & 31;
  const int wave = tid >> 5;
  const int m    = lane & 15;
  const int half = lane >> 4;
  const int wr   = wave & 3;   // tile row (0..3)
  const int wp   = wave >> 2;  // tile-col pair (0..1)

  const int rm0 = blockIdx.x * TILE;
  const int cn0 = blockIdx.y * TILE;

  const uint64_t fbase   = (uint64_t)(uintptr_t)F;
  const uint32_t aLds0   = (uint32_t)(uintptr_t)&As[0][0][0];
  const uint32_t bLds0   = (uint32_t)(uintptr_t)&Bs[0][0][0];
  const uint32_t bufSz   = (uint32_t)(TILE * LDSP * sizeof(float));

  // Stage one 64(rows)x64(K) tile of A and of B: 2 x 1024 16B chunks,
  // 8 async b128 issues per thread per chunk (ASYNCcnt-tracked).
  auto issue_chunk = [&](int buf, int k0) {
#pragma unroll
    for (int q = 0; q < 4; ++q) {
      const int id  = tid + 256 * q;
      const int row = id >> 4;          // 0..63
      const int cc  = (id & 15) << 2;   // float column of the 16B chunk
      const uint32_t goffA = (uint32_t)(((rm0 + row) * D + k0 + cc) * sizeof(float));
      const uint32_t goffB = (uint32_t)(((cn0 + row) * D + k0 + cc) * sizeof(float));
      const uint32_t lA = aLds0 + (uint32_t)buf * bufSz + (uint32_t)(row * LDSP + cc) * sizeof(float);
      const uint32_t lB = bLds0 + (uint32_t)buf * bufSz + (uint32_t)(row * LDSP + cc) * sizeof(float);
      asm volatile("global_load_async_to_lds_b128 %0, %1, %2"
                   :: "v"(lA), "v"(goffA), "s"(fbase) : "memory");
      asm volatile("global_load_async_to_lds_b128 %0, %1, %2"
                   :: "v"(lB), "v"(goffB), "s"(fbase) : "memory");
    }
  };

  v8f acc0 = {}, acc1 = {};

  issue_chunk(0, 0);
  asm volatile("s_wait_asynccnt 0" ::: "memory");
  __syncthreads();

  const int nch = D / KCH;  // 32
  int buf = 0;
  for (int c = 0; c < nch; ++c) {
    // Prefetch next chunk into the other buffer (consumed+barriered last iter).
    if (c + 1 < nch) issue_chunk(buf ^ 1, (c + 1) * KCH);

    const float* arow = &As[buf][16 * wr + m][2 * half];
    const float* b0r  = &Bs[buf][16 * (2 * wp + 0) + m][2 * half];
    const float* b1r  = &Bs[buf][16 * (2 * wp + 1) + m][2 * half];
#pragma unroll
    for (int kk = 0; kk < KCH; kk += 4) {
      v2f a  = *(const v2f*)(arow + kk);
      v2f b0 = *(const v2f*)(b0r + kk);
      v2f b1 = *(const v2f*)(b1r + kk);
      acc0 = __builtin_amdgcn_wmma_f32_16x16x4_f32(false, a, false, b0,
                                                   (short)0, acc0, false, false);
      acc1 = __builtin_amdgcn_wmma_f32_16x16x4_f32(false, a, false, b1,
                                                   (short)0, acc1, false, false);
    }

    // Publish the prefetched buffer: own async ops done, then block barrier.
    if (c + 1 < nch) { asm volatile("s_wait_asynccnt 0" ::: "memory"); }
    __syncthreads();
    buf ^= 1;
  }

  // C/D layout: lane 0-15 -> N=lane, M=v; lane 16-31 -> M=v+8.
  const int gr  = rm0 + 16 * wr + half * 8;
  const int gc0 = cn0 + 32 * wp + m;
#pragma unroll
  for (int v = 0; v < 8; ++v) {
    G[(size_t)(gr + v) * N + gc0]      = acc0[v];
    G[(size_t)(gr + v) * N + gc0 + 16] = acc1[v];
  }
}

// ---------------------------------------------------------------------------
// Deterministic block tree-reduction (blockDim.x == 256).
// ---------------------------------------------------------------------------
__device__ __forceinline__ float block_reduce_sum256(float v, float* red) {
  const int tid = threadIdx.x;
  red[tid] = v;
  __syncthreads();
  for (int s = 128; s > 0; s >>= 1) {
    if (tid < s) red[tid] += red[tid + s];
    __syncthreads();
  }
  float r = red[0];
  __syncthreads();
  return r;
}

// Row-wise softmax of G[row, cols[0..cnt)] restricted to the mask columns.
__device__ __forceinline__ void masked_softmax_row(const float* __restrict__ G, int N,
                                                   int row, const int* __restrict__ cols,
                                                   int cnt, float* __restrict__ dst) {
  float mx = -INFINITY;
  for (int k = 0; k < cnt; ++k) mx = fmaxf(mx, G[(size_t)row * N + cols[k]]);
  float s = 0.0f;
  for (int k = 0; k < cnt; ++k) {
    float e = __expf(G[(size_t)row * N + cols[k]] - mx);
    dst[k] = e;
    s += e;
  }
  const float inv = 1.0f / s;
  for (int k = 0; k < cnt; ++k) dst[k] *= inv;
}

// ---------------------------------------------------------------------------
// Kernel 2: per-anchor transition loss. One block (256 threads) per anchor.
// ---------------------------------------------------------------------------
__global__ __launch_bounds__(256) void anchor_loss_kernel(const float* __restrict__ G,
                                                          const int* __restrict__ targets,
                                                          float* __restrict__ lossOut,
                                                          int N) {
  __shared__ float Asm[MAXM * MAXM];    // 64 KB: softmax rows over col-mask (ab)
  __shared__ float Bsm[MAXM * MAXM];    // 64 KB: softmax rows over row-mask (ba)
  __shared__ int idx_ni[MAXM], idx_nc[MAXM];
  __shared__ int idx_pi[MAXP], idx_pc[MAXP];
  __shared__ int cnt4[4];
  __shared__ float red[256];

  const int i   = blockIdx.x;
  const int tid = threadIdx.x;
  const int ti  = targets[i];
  const int half = N >> 1;
  const int si   = (i >= half) ? 1 : 0;

  // Build index lists deterministically (single thread, N=256 scan).
  if (tid == 0) {
    int cni = 0, cnc = 0, cpi = 0, cpc = 0;
    for (int j = 0; j < N; ++j) {
      const bool pos   = (targets[j] == ti);
      const bool intra = (((j >= half) ? 1 : 0) == si);
      if (pos) {
        if (intra) { if (cpi < MAXP) idx_pi[cpi++] = j; }
        else       { if (cpc < MAXP) idx_pc[cpc++] = j; }
      } else {
        if (intra) { if (cni < MAXM) idx_ni[cni++] = j; }
        else       { if (cnc < MAXM) idx_nc[cnc++] = j; }
      }
    }
    cnt4[0] = cni; cnt4[1] = cnc; cnt4[2] = cpi; cnt4[3] = cpc;
  }
  __syncthreads();
  const int cni = cnt4[0], cnc = cnt4[1], cpi = cnt4[2], cpc = cnt4[3];

  // ---- negative branch: t_neg on (mni x mni) via msm(mnc) @ msm(mni) ----
  if (tid < cni) {                       // ab rows: r in mni, softmax over mnc
    masked_softmax_row(G, N, idx_ni[tid], idx_nc, cnc, &Asm[tid * MAXM]);
  } else if (tid >= 128 && tid < 128 + cnc) {  // ba rows: k in mnc, softmax over mni
    const int kk = tid - 128;
    masked_softmax_row(G, N, idx_nc[kk], idx_ni, cni, &Bsm[kk * MAXM]);
  }
  __syncthreads();

  float accum = 0.0f;
  const int nPairsN = cni * cni;
  for (int p = tid; p < nPairsN; p += 256) {
    const int r = p / cni;
    const int c = p - r * cni;
    float t = 0.0f;
    const float* ar = &Asm[r * MAXM];
    for (int k = 0; k < cnc; ++k) t += ar[k] * Bsm[k * MAXM + c];
    const float tv = sqrtf(fmaxf(t, EPS_T));
    accum += fmaxf(tv + MARGIN_MINUS_ALPHA, 0.0f);
  }
  const float sumN = block_reduce_sum256(accum, red);
  const float loss_ap = (cni > 0) ? (sumN / (float)(cni * cni)) : 0.0f;
  __syncthreads();   // Asm/Bsm reused below

  // ---- positive branch: t_pos on (mpi x mpi) via msm(mpc) @ msm(mpi) ----
  if (tid < cpi) {
    masked_softmax_row(G, N, idx_pi[tid], idx_pc, cpc, &Asm[tid * MAXM]);
  } else if (tid >= 128 && tid < 128 + cpc) {
    const int kk = tid - 128;
    masked_softmax_row(G, N, idx_pc[kk], idx_pi, cpi, &Bsm[kk * MAXM]);
  }
  __syncthreads();

  float accumP = 0.0f;
  const int nPairsP = cpi * cpi;
  for (int p = tid; p < nPairsP; p += 256) {
    const int r = p / cpi;
    const int c = p - r * cpi;
    float t = 0.0f;
    const float* ar = &Asm[r * MAXM];
    for (int k = 0; k < cpc; ++k) t += ar[k] * Bsm[k * MAXM + c];
    const float tv = sqrtf(fmaxf(t, EPS_T));
    accumP += fmaxf(ALPHA_1F - tv, 0.0f);
  }
  const float sumP = block_reduce_sum256(accumP, red);
  const float loss_an = (cpi > 0) ? (sumP / (float)(cpi * cpi)) : 0.0f;

  if (tid == 0) lossOut[i] = loss_ap + loss_an;
}

// ---------------------------------------------------------------------------
// Kernel 3: hard-mining accuracy. dist^2(i,j) = Gii + Gjj - 2*Gij.
// ---------------------------------------------------------------------------
__global__ __launch_bounds__(256) void accuracy_kernel(const float* __restrict__ G,
                                                       const int* __restrict__ targets,
                                                       float* __restrict__ corr,
                                                       int N) {
  __shared__ float smax[256];
  __shared__ float smin[256];
  const int i   = blockIdx.x;
  const int tid = threadIdx.x;
  const int ti  = targets[i];
  const float gii = G[(size_t)i * N + i];

  float dap = -INFINITY, dan = INFINITY;
  for (int j = tid; j < N; j += blockDim.x) {
    const float d2 = gii + G[(size_t)j * N + j] - 2.0f * G[(size_t)i * N + j];
    const float d  = sqrtf(fmaxf(d2, EPS_T));
    if (targets[j] == ti) dap = fmaxf(dap, d);
    else                  dan = fminf(dan, d);
  }
  smax[tid] = dap; smin[tid] = dan;
  __syncthreads();
  for (int s = 128; s > 0; s >>= 1) {
    if (tid < s) {
      smax[tid] = fmaxf(smax[tid], smax[tid + s]);
      smin[tid] = fminf(smin[tid], smin[tid + s]);
    }
    __syncthreads();
  }
  if (tid == 0) corr[i] = (smin[0] >= smax[0]) ? 1.0f : 0.0f;
}

// ---------------------------------------------------------------------------
// Kernel 4: final reductions.
// ---------------------------------------------------------------------------
__global__ __launch_bounds__(256) void finalize_kernel(const float* __restrict__ lossArr,
                                                       const float* __restrict__ corrArr,
                                                       float* __restrict__ out, int N) {
  __shared__ float r0[256];
  __shared__ float r1[256];
  const int tid = threadIdx.x;
  float s0 = 0.0f, s1 = 0.0f;
  for (int j = tid; j < N; j += 256) { s0 += lossArr[j]; s1 += corrArr[j]; }
  r0[tid] = s0; r1[tid] = s1;
  __syncthreads();
  for (int s = 128; s > 0; s >>= 1) {
    if (tid < s) { r0[tid] += r0[tid + s]; r1[tid] += r1[tid + s]; }
    __syncthreads();
  }
  if (tid == 0) {
    out[0] = r0[0] / (float)N;  // loss (mean over anchors)
    out[1] = r1[0];             // correct count, as float per output convention
  }
}

// ---------------------------------------------------------------------------
extern "C" void kernel_launch(void* const* d_in, const int* in_sizes, int n_in,
                              void* d_out, int out_size, void* d_ws, size_t ws_size,
                              hipStream_t stream) {
  (void)n_in; (void)out_size; (void)ws_size;
  const float* F       = (const float*)d_in[0];
  const int*   targets = (const int*)d_in[1];
  const int N = in_sizes[1];         // 256
  const int D = in_sizes[0] / N;     // 2048

  float* G       = (float*)d_ws;               // N*N f32
  float* lossArr = G + (size_t)N * N;          // N f32
  float* corrArr = lossArr + N;                // N f32

  dim3 gGrid(N / TILE, N / TILE);
  gram_wmma_kernel<<<gGrid, 256, 0, stream>>>(F, G, N, D);
  anchor_loss_kernel<<<N, 256, 0, stream>>>(G, targets, lossArr, N);
  accuracy_kernel<<<N, 256, 0, stream>>>(G, targets, corrArr, N);
  finalize_kernel<<<1, 256, 0, stream>>>(lossArr, corrArr, (float*)d_out, N);
}